// PVAM_59596966199749
// MI455X (gfx1250) — compile-verified
//
#include <hip/hip_runtime.h>
#include <hip/hip_bf16.h>
#include <math.h>

typedef __attribute__((ext_vector_type(2))) float v2f;
typedef __attribute__((ext_vector_type(8))) float v8f;

#define B_DIM 64
#define N_DIM 256
#define HW    256   // 8*32

__device__ __forceinline__ float fast_tanh(float x) {
#if __has_builtin(__builtin_amdgcn_tanhf)
    return __builtin_amdgcn_tanhf(x);   // gfx1250 v_tanh_f32 trans op
#else
    return tanhf(x);
#endif
}

// ---------------------------------------------------------------------------
// K1: att_r[t, d] = sum_k pe[t,k] * wo_w[d,k] + wo_b[d]     (T x 256)
// pe = standard sincos positional embedding, computed on the fly into LDS.
// Single block, 256 threads (thread == output column d).
// ---------------------------------------------------------------------------
__global__ __launch_bounds__(256) void pvam_attr_kernel(
    const float* __restrict__ wo_w, const float* __restrict__ wo_b,
    float* __restrict__ attr, int T) {
    __shared__ float pe[32 * N_DIM];
    const int d = threadIdx.x;
    // inv_freq for this column: 10000^(-(2*(d/2))/D) = exp(-2i * ln(1e4)/D)
    const int i2 = (d >> 1) * 2;
    const float inv = __expf(-(float)i2 * (9.210340371976184f / (float)N_DIM));
    const float bias = wo_b[d];
    const float* wrow = wo_w + d * N_DIM;

    for (int t0 = 0; t0 < T; t0 += 32) {
        // build 32 rows of pe (zero-padded past T so fixed-trip loops are safe)
        #pragma unroll
        for (int tt = 0; tt < 32; ++tt) {
            int t = t0 + tt;
            float ang = (float)t * inv;
            float v = (d & 1) ? cosf(ang) : sinf(ang);
            pe[tt * N_DIM + d] = (t < T) ? v : 0.0f;
        }
        __syncthreads();

        float acc[32];
        #pragma unroll
        for (int tt = 0; tt < 32; ++tt) acc[tt] = 0.0f;

        for (int k = 0; k < N_DIM; ++k) {
            float w = wrow[k];
            #pragma unroll
            for (int tt = 0; tt < 32; ++tt)
                acc[tt] = fmaf(pe[tt * N_DIM + k], w, acc[tt]);
        }

        #pragma unroll
        for (int tt = 0; tt < 32; ++tt) {
            int t = t0 + tt;
            if (t < T) attr[t * N_DIM + d] = acc[tt] + bias;
        }
        __syncthreads();
    }
}

// ---------------------------------------------------------------------------
// K2: att_x = xh @ wv_w^T + wv_b  via V_WMMA_F32_16X16X4_F32
//   A[m][k] = x[b*65536 + k*256 + s]   (m = b*HW + s; transpose of x's layout)
//   B[k][n] = wv_w[n*256 + k]
// Block = 4 waves (128 thr), block tile 64M x 64N, wave tile 16M x 64N.
// Grid  = (16384/64, 256/64) = (256, 4).
// ---------------------------------------------------------------------------
__global__ __launch_bounds__(128) void pvam_attx_gemm(
    const float* __restrict__ x, const float* __restrict__ wv_w,
    const float* __restrict__ wv_b, float* __restrict__ attx) {
    const int lane = threadIdx.x & 31;
    const int wave = threadIdx.x >> 5;
    const int half = lane >> 4;        // 0: K={0,1}  1: K={2,3}
    const int lidx = lane & 15;
    const int kk   = half * 2;

    const int m0 = blockIdx.x * 64 + wave * 16;
    const int n0 = blockIdx.y * 64;

    // A fragment addressing (strided: element k lives at aBase[k*HW])
    const int m  = m0 + lidx;
    const int bb = m >> 8;             // m / HW
    const int ss = m & 255;            // m % HW
    const float* aBase = x + bb * (N_DIM * HW) + ss;

    // B fragment rows (contiguous in k -> 8-byte vector loads)
    const float* br0 = wv_w + (n0 +  0 + lidx) * N_DIM;
    const float* br1 = wv_w + (n0 + 16 + lidx) * N_DIM;
    const float* br2 = wv_w + (n0 + 32 + lidx) * N_DIM;
    const float* br3 = wv_w + (n0 + 48 + lidx) * N_DIM;

    v8f c0 = {}, c1 = {}, c2 = {}, c3 = {};

    for (int k0 = 0; k0 < N_DIM; k0 += 4) {
        const int k = k0 + kk;
        v2f a;
        a.x = aBase[k * HW];
        a.y = aBase[(k + 1) * HW];
        v2f f0 = *(const v2f*)(br0 + k);
        v2f f1 = *(const v2f*)(br1 + k);
        v2f f2 = *(const v2f*)(br2 + k);
        v2f f3 = *(const v2f*)(br3 + k);
        c0 = __builtin_amdgcn_wmma_f32_16x16x4_f32(false, a, false, f0, (short)0, c0, false, false);
        c1 = __builtin_amdgcn_wmma_f32_16x16x4_f32(false, a, false, f1, (short)0, c1, false, false);
        c2 = __builtin_amdgcn_wmma_f32_16x16x4_f32(false, a, false, f2, (short)0, c2, false, false);
        c3 = __builtin_amdgcn_wmma_f32_16x16x4_f32(false, a, false, f3, (short)0, c3, false, false);
    }

    // Epilogue: C layout -> VGPR v holds M = m0 + v + half*8, N = n0' + lidx
    const float bv0 = wv_b[n0 +  0 + lidx];
    const float bv1 = wv_b[n0 + 16 + lidx];
    const float bv2 = wv_b[n0 + 32 + lidx];
    const float bv3 = wv_b[n0 + 48 + lidx];
    #pragma unroll
    for (int v = 0; v < 8; ++v) {
        const int M = m0 + v + half * 8;
        float* orow = attx + M * N_DIM;
        orow[n0 +  0 + lidx] = c0[v] + bv0;
        orow[n0 + 16 + lidx] = c1[v] + bv1;
        orow[n0 + 32 + lidx] = c2[v] + bv2;
        orow[n0 + 48 + lidx] = c3[v] + bv3;
    }
}

// ---------------------------------------------------------------------------
// K3: fused score -> softmax -> weighted sum, one block per (t, b).
// Phase 1: thread = spatial s:  score[s] = sum_d tanh(ax[s,d]+ar[d])*we[d]+we_b
// Phase 2: thread = channel d:  out[d]   = sum_s alpha[s]*ax[s,d] / HW
// ---------------------------------------------------------------------------
__global__ __launch_bounds__(256) void pvam_att_fused(
    const float* __restrict__ attx, const float* __restrict__ attr,
    const float* __restrict__ we_w, const float* __restrict__ we_b,
    float* __restrict__ out, int T) {
    __shared__ float s_ar[N_DIM];
    __shared__ float s_we[N_DIM];
    __shared__ float s_red[HW];
    __shared__ float s_alpha[HW];

    const int tid = threadIdx.x;
    const int t = blockIdx.x;
    const int b = blockIdx.y;

    s_ar[tid] = attr[t * N_DIM + tid];
    s_we[tid] = we_w[tid];
    __syncthreads();

    // -------- phase 1: additive attention score for s = tid --------
    const float* axrow = attx + b * (HW * N_DIM) + tid * N_DIM;
    float acc = 0.0f;
    #pragma unroll 4
    for (int d = 0; d < N_DIM; d += 4) {
        float4 v = *(const float4*)(axrow + d);
        acc = fmaf(fast_tanh(v.x + s_ar[d + 0]), s_we[d + 0], acc);
        acc = fmaf(fast_tanh(v.y + s_ar[d + 1]), s_we[d + 1], acc);
        acc = fmaf(fast_tanh(v.z + s_ar[d + 2]), s_we[d + 2], acc);
        acc = fmaf(fast_tanh(v.w + s_ar[d + 3]), s_we[d + 3], acc);
    }
    const float score = acc + we_b[0];

    // -------- softmax over the 256 spatial positions --------
    s_red[tid] = score;
    __syncthreads();
    for (int off = 128; off > 0; off >>= 1) {
        if (tid < off) s_red[tid] = fmaxf(s_red[tid], s_red[tid + off]);
        __syncthreads();
    }
    const float mx = s_red[0];
    __syncthreads();

    const float e = __expf(score - mx);
    s_red[tid] = e;
    __syncthreads();
    for (int off = 128; off > 0; off >>= 1) {
        if (tid < off) s_red[tid] += s_red[tid + off];
        __syncthreads();
    }
    const float denom = s_red[0];
    s_alpha[tid] = e / denom;
    __syncthreads();

    // -------- phase 2: out[b,t,d] for d = tid (coalesced over d) --------
    const float* axb = attx + b * (HW * N_DIM);
    float o = 0.0f;
    #pragma unroll 4
    for (int s = 0; s < HW; ++s)
        o = fmaf(s_alpha[s], axb[s * N_DIM + tid], o);

    out[(b * T + t) * N_DIM + tid] = o * (1.0f / (float)HW);
}

extern "C" void kernel_launch(void* const* d_in, const int* in_sizes, int n_in,
                              void* d_out, int out_size, void* d_ws, size_t ws_size,
                              hipStream_t stream) {
    const float* x    = (const float*)d_in[0];
    // d_in[1] = seq_len (device int; unreadable during graph capture)
    const float* wo_w = (const float*)d_in[2];
    const float* wo_b = (const float*)d_in[3];
    const float* wv_w = (const float*)d_in[4];
    const float* wv_b = (const float*)d_in[5];
    const float* we_w = (const float*)d_in[6];
    const float* we_b = (const float*)d_in[7];
    float* out = (float*)d_out;

    const int T = out_size / (B_DIM * N_DIM);   // = seq_len

    float* attx = (float*)d_ws;                          // 16384 * 256 f32
    float* attr = attx + (B_DIM * HW) * N_DIM;           // T * 256 f32

    pvam_attr_kernel<<<1, 256, 0, stream>>>(wo_w, wo_b, attr, T);
    pvam_attx_gemm<<<dim3((B_DIM * HW) / 64, N_DIM / 64), 128, 0, stream>>>(
        x, wv_w, wv_b, attx);
    pvam_att_fused<<<dim3(T, B_DIM), 256, 0, stream>>>(attx, attr, we_w, we_b, out, T);
}